// DynamicGraphConv_4389456577118
// MI455X (gfx1250) — compile-verified
//
#include <hip/hip_runtime.h>

#define NN   4096   // NUM_NODES
#define EMB  10
#define BB   64     // BATCH
#define HH   64     // INPUT_DIM
#define OO   64     // OUTPUT_DIM
#define CATD 448    // 7*64

typedef __attribute__((ext_vector_type(2))) float v2f;
typedef __attribute__((ext_vector_type(8))) float v8f;

// ---------------------------------------------------------------------------
// Kernel 1: row sums of the two support matrices.  grid = (NN, 2), block = 256
// ---------------------------------------------------------------------------
__global__ void __launch_bounds__(256)
rowsum_kernel(const float* __restrict__ sup, float* __restrict__ r) {
  const int n = blockIdx.x;
  const int s = blockIdx.y;
  const float* row = sup + ((size_t)s * NN + n) * NN;
  float acc = 0.f;
  for (int m = threadIdx.x; m < NN; m += 256) acc += row[m];
  __shared__ float red[256];
  red[threadIdx.x] = acc;
  __syncthreads();
  for (int off = 128; off > 0; off >>= 1) {
    if ((int)threadIdx.x < off) red[threadIdx.x] += red[threadIdx.x + off];
    __syncthreads();
  }
  if (threadIdx.x == 0) r[(size_t)s * NN + n] = red[0];
}

// ---------------------------------------------------------------------------
// Kernel 2: r3[n] = sum_m softmax(relu(E1 @ E2), axis=1)[n,m]  (faithful ~1.0)
// grid = NN, block = 256; each thread owns 16 columns.
// ---------------------------------------------------------------------------
__global__ void __launch_bounds__(256)
adaptive_rowsum_kernel(const float* __restrict__ E1, const float* __restrict__ E2,
                       float* __restrict__ r3) {
  const int n   = blockIdx.x;
  const int tid = threadIdx.x;
  __shared__ float e1[16];
  __shared__ float red[256];
  if (tid < EMB) e1[tid] = E1[n * EMB + tid];
  __syncthreads();

  float z[16];
  float mx = 0.f;  // relu => z >= 0
  #pragma unroll
  for (int i = 0; i < 16; ++i) {
    const int m = tid + i * 256;
    float d = 0.f;
    #pragma unroll
    for (int e = 0; e < EMB; ++e) d = fmaf(e1[e], E2[e * NN + m], d);
    d = d > 0.f ? d : 0.f;
    z[i] = d;
    mx = fmaxf(mx, d);
  }
  // block max
  red[tid] = mx; __syncthreads();
  for (int off = 128; off > 0; off >>= 1) {
    if (tid < off) red[tid] = fmaxf(red[tid], red[tid + off]);
    __syncthreads();
  }
  const float M = red[0]; __syncthreads();
  // block sum of exp(z - M)
  float sl = 0.f;
  #pragma unroll
  for (int i = 0; i < 16; ++i) sl += __expf(z[i] - M);
  red[tid] = sl; __syncthreads();
  for (int off = 128; off > 0; off >>= 1) {
    if (tid < off) red[tid] += red[tid + off];
    __syncthreads();
  }
  const float S = red[0]; __syncthreads();
  // block sum of softmax entries
  float q = 0.f;
  #pragma unroll
  for (int i = 0; i < 16; ++i) q += __expf(z[i] - M) / S;
  red[tid] = q; __syncthreads();
  for (int off = 128; off > 0; off >>= 1) {
    if (tid < off) red[tid] += red[tid + off];
    __syncthreads();
  }
  if (tid == 0) r3[n] = red[0];
}

// ---------------------------------------------------------------------------
// Kernel 3: per-node folded-weight GEMM via V_WMMA_F32_16X16X4_F32.
// One block per node, 128 threads = 4 wave32s.
//   Weff[o][h] = sum_j s_j * W[o, j*64+h]   (built in LDS)
//   out[b,n,o] = sum_h x[b,n,h] * Weff[o][h] + bias[o]
// Wave w computes batch rows 16w..16w+15 for all 64 outputs (4 16x16 tiles).
// ---------------------------------------------------------------------------
__global__ void __launch_bounds__(128)
gemm_node_kernel(const float* __restrict__ x, const float* __restrict__ W,
                 const float* __restrict__ bias, const float* __restrict__ rws,
                 float* __restrict__ out) {
  const int n   = blockIdx.x;
  const int tid = threadIdx.x;

  __shared__ float Xs[64][65];   // padded: conflict-free column reads
  __shared__ float Ws[64][65];   // Weff[o][h]

  const float r1 = rws[n];
  const float r2 = rws[NN + n];
  const float r3 = rws[2 * NN + n];
  const float s[7] = {1.f, r1, r1 * r1, r2, r2 * r2, r3, r3 * r3};

  // Stage X_n tile (64 batch rows x 64 features), coalesced 256B per row.
  for (int i = tid; i < 64 * 64; i += 128) {
    const int b = i >> 6, h = i & 63;
    Xs[b][h] = x[((size_t)b * NN + n) * HH + h];
  }
  // Fold the 7 per-node scales into one 64x64 effective weight (W is L2-resident).
  for (int i = tid; i < 64 * 64; i += 128) {
    const int o = i >> 6, h = i & 63;
    const float* wrow = W + (size_t)o * CATD + h;
    float acc = 0.f;
    #pragma unroll
    for (int j = 0; j < 7; ++j) acc = fmaf(s[j], wrow[j * 64], acc);
    Ws[o][h] = acc;
  }
  __syncthreads();

  const int lane  = tid & 31;
  const int wave  = tid >> 5;
  const int m0    = wave * 16;          // batch-row base for this wave
  const int mrow  = m0 + (lane & 15);   // A: lanes 0-15 / 16-31 both cover M=0..15
  const int col   = lane & 15;          // B/C/D: N = lane % 16
  const int khalf = (lane >> 4) * 2;    // lanes 16-31 hold K+2,K+3

  v8f c0 = {}, c1 = {}, c2 = {}, c3 = {};

  #pragma unroll
  for (int k = 0; k < HH; k += 4) {
    v2f a;                       // A[m, k..k+3], 16x4 f32 layout
    a.x = Xs[mrow][k + khalf];
    a.y = Xs[mrow][k + khalf + 1];
    v2f b0, b1, b2, b3;          // B[k][o] = Weff[o][k]  (4x16 f32 layout)
    b0.x = Ws[col][k + khalf];      b0.y = Ws[col][k + khalf + 1];
    b1.x = Ws[16 + col][k + khalf]; b1.y = Ws[16 + col][k + khalf + 1];
    b2.x = Ws[32 + col][k + khalf]; b2.y = Ws[32 + col][k + khalf + 1];
    b3.x = Ws[48 + col][k + khalf]; b3.y = Ws[48 + col][k + khalf + 1];
    c0 = __builtin_amdgcn_wmma_f32_16x16x4_f32(false, a, false, b0, (short)0, c0, false, false);
    c1 = __builtin_amdgcn_wmma_f32_16x16x4_f32(false, a, false, b1, (short)0, c1, false, false);
    c2 = __builtin_amdgcn_wmma_f32_16x16x4_f32(false, a, false, b2, (short)0, c2, false, false);
    c3 = __builtin_amdgcn_wmma_f32_16x16x4_f32(false, a, false, b3, (short)0, c3, false, false);
  }

  // D layout: VGPR r -> M = r (lanes 0-15) / M = 8+r (lanes 16-31); N = lane%16.
  const float bo0 = bias[col];
  const float bo1 = bias[16 + col];
  const float bo2 = bias[32 + col];
  const float bo3 = bias[48 + col];
  #pragma unroll
  for (int r = 0; r < 8; ++r) {
    const int b = m0 + r + 8 * (lane >> 4);
    float* orow = out + ((size_t)b * NN + n) * OO;
    orow[col]      = c0[r] + bo0;
    orow[16 + col] = c1[r] + bo1;
    orow[32 + col] = c2[r] + bo2;
    orow[48 + col] = c3[r] + bo3;
  }
}

// ---------------------------------------------------------------------------
extern "C" void kernel_launch(void* const* d_in, const int* in_sizes, int n_in,
                              void* d_out, int out_size, void* d_ws, size_t ws_size,
                              hipStream_t stream) {
  const float* x    = (const float*)d_in[0];  // [64, 4096, 64]
  const float* sup  = (const float*)d_in[1];  // [2, 4096, 4096]
  const float* nv1  = (const float*)d_in[2];  // [4096, 10]
  const float* nv2  = (const float*)d_in[3];  // [10, 4096]
  const float* W    = (const float*)d_in[4];  // [64, 448]
  const float* bias = (const float*)d_in[5];  // [64]
  float* out = (float*)d_out;
  float* rws = (float*)d_ws;                  // 3*NN floats: r1 | r2 | r3

  dim3 g1(NN, 2);
  rowsum_kernel<<<g1, 256, 0, stream>>>(sup, rws);
  adaptive_rowsum_kernel<<<NN, 256, 0, stream>>>(nv1, nv2, rws + 2 * NN);
  gemm_node_kernel<<<NN, 128, 0, stream>>>(x, W, bias, rws, out);
}